// GCNModifiedAEBoostingFirstLayer_72756745994510
// MI455X (gfx1250) — compile-verified
//
#include <hip/hip_runtime.h>
#include <hip/hip_bf16.h>

// CDNA5 (gfx1250) f32 WMMA GEMM pipeline for the GCN autoencoder reference.
// v_wmma_f32_16x16x4_f32 compute + async global->LDS staging (ASYNCcnt) with
// double buffering. wave32, 8 waves/block, 128x128 block tile, BK=16.

typedef __attribute__((ext_vector_type(2))) float v2f;
typedef __attribute__((ext_vector_type(8))) float v8f;

#define TILE_M 128
#define TILE_N 128
#define TILE_K 16
#define AS_STRIDE 20        // floats; 80B row stride (16B-aligned for b128 async writes)
#define BS_PAIR_STRIDE 288  // floats per k-pair row; 288 % 64 == 32 -> hi half-wave on other banks

// ---- gfx1250 async global->LDS DMA (tracked by ASYNCcnt) ----
// Low 32 bits of a generic pointer into LDS == byte offset in the group segment
// (LDS aperture truncates to addr[31:0], ISA 10.2), which is exactly what the
// VDST operand of global_load_async_to_lds_* wants.
__device__ __forceinline__ void async_b128(const float* g, const float* l) {
    asm volatile("global_load_async_to_lds_b128 %0, %1, off"
                 :: "v"((uint32_t)(uintptr_t)l), "v"((uint64_t)(uintptr_t)g)
                 : "memory");
}
__device__ __forceinline__ void async_b64(const float* g, const float* l) {
    asm volatile("global_load_async_to_lds_b64 %0, %1, off"
                 :: "v"((uint32_t)(uintptr_t)l), "v"((uint64_t)(uintptr_t)g)
                 : "memory");
}
__device__ __forceinline__ void async_b32(const float* g, const float* l) {
    asm volatile("global_load_async_to_lds_b32 %0, %1, off"
                 :: "v"((uint32_t)(uintptr_t)l), "v"((uint64_t)(uintptr_t)g)
                 : "memory");
}
template <int Nwait> __device__ __forceinline__ void wait_async() {
    if constexpr (Nwait == 0)      asm volatile("s_wait_asynccnt 0" ::: "memory");
    else if constexpr (Nwait == 6) asm volatile("s_wait_asynccnt 6" ::: "memory");
    else                           asm volatile("s_wait_asynccnt 10" ::: "memory");
}

// C = A(MxK) * B(KxN) row-major; TRANSB: B is (N x K) accessed transposed;
// ZERO_DIAG zeroes C's diagonal (two-hop adjacency). Requires K % 16 == 0.
template <bool TRANSB, bool ZERO_DIAG>
__global__ __launch_bounds__(256, 2) void wmma_gemm_f32(
    const float* __restrict__ A, const float* __restrict__ B, float* __restrict__ C,
    int M, int Nc, int K, int lda, int ldb, int ldc)
{
    __shared__ float As[2][TILE_M * AS_STRIDE];
    __shared__ float Bs[2][(TILE_K / 2) * BS_PAIR_STRIDE];

    const int tid  = threadIdx.x;
    const int lane = tid & 31;
    const int hi   = lane >> 4;  // half-wave select
    const int l15  = lane & 15;
    const int wave = tid >> 5;

    const int wm0 = (wave & 3) * 32;   // wave sub-tile origin (M)
    const int wn0 = (wave >> 2) * 64;  // wave sub-tile origin (N)

    const int bm0 = blockIdx.y * TILE_M;
    const int bn0 = blockIdx.x * TILE_N;

    // issue one K-tile's async loads (uniform EXEC; edge rows/cols clamped,
    // padding garbage only feeds masked-out C entries since K % 16 == 0)
    auto stage = [&](int kt, int buf) {
#pragma unroll
        for (int i = 0; i < 2; ++i) {            // A: 128x16, 2 x b128 per thread
            int c = tid + i * 256;               // 0..511
            int row = c >> 2, col = (c & 3) << 2;
            int grow = bm0 + row; grow = grow < M ? grow : M - 1;
            async_b128(A + (size_t)grow * lda + kt + col,
                       &As[buf][row * AS_STRIDE + col]);
        }
        if constexpr (!TRANSB) {                 // B: 16x128, pair-interleaved in LDS
#pragma unroll
            for (int i = 0; i < 8; ++i) {        // 8 x b32 per thread
                int c = tid + i * 256;           // 0..2047
                int k = c >> 7, n = c & 127;
                int gn = bn0 + n; gn = gn < Nc ? gn : Nc - 1;
                async_b32(B + (size_t)(kt + k) * ldb + gn,
                          &Bs[buf][(k >> 1) * BS_PAIR_STRIDE + 2 * n + (k & 1)]);
            }
        } else {                                 // B is N x K: k-pair contiguous -> b64
#pragma unroll
            for (int i = 0; i < 4; ++i) {        // 4 x b64 per thread
                int c = tid + i * 256;           // 0..1023
                int j = c >> 7, n = c & 127;     // j = k-pair index 0..7
                int gn = bn0 + n; gn = gn < Nc ? gn : Nc - 1;
                async_b64(B + (size_t)gn * ldb + kt + 2 * j,
                          &Bs[buf][j * BS_PAIR_STRIDE + 2 * n]);
            }
        }
    };
    constexpr int PER_TILE = TRANSB ? 6 : 10;    // async instrs per wave per tile

    v8f acc[2][4];
#pragma unroll
    for (int mi = 0; mi < 2; ++mi)
#pragma unroll
        for (int ni = 0; ni < 4; ++ni)
#pragma unroll
            for (int r = 0; r < 8; ++r) acc[mi][ni][r] = 0.0f;

    const int ntiles = K / TILE_K;
    stage(0, 0);

    for (int idx = 0; idx < ntiles; ++idx) {
        const int cur = idx & 1;
        if (idx + 1 < ntiles) {
            stage((idx + 1) * TILE_K, cur ^ 1);
            wait_async<PER_TILE>();   // newest batch may be in flight; cur's is done
        } else {
            wait_async<0>();
        }
        __syncthreads();              // all waves' cur-tile DMA landed in LDS

        // 4 k-steps of 16x16x4; fragments are single aligned ds_load_b64 each
#pragma unroll
        for (int ks = 0; ks < TILE_K; ks += 4) {
            v2f afrag[2], bfrag[4];
#pragma unroll
            for (int mi = 0; mi < 2; ++mi) {
                int m = wm0 + mi * 16 + l15;     // A frag: K = ks+2*hi, ks+2*hi+1
                afrag[mi] = *reinterpret_cast<const v2f*>(
                    &As[cur][m * AS_STRIDE + ks + 2 * hi]);
            }
#pragma unroll
            for (int ni = 0; ni < 4; ++ni) {
                int n = wn0 + ni * 16 + l15;     // B frag: pair (ks/2 + hi) at col n
                bfrag[ni] = *reinterpret_cast<const v2f*>(
                    &Bs[cur][(ks / 2 + hi) * BS_PAIR_STRIDE + 2 * n]);
            }
#pragma unroll
            for (int mi = 0; mi < 2; ++mi)
#pragma unroll
                for (int ni = 0; ni < 4; ++ni)
                    acc[mi][ni] = __builtin_amdgcn_wmma_f32_16x16x4_f32(
                        false, afrag[mi], false, bfrag[ni],
                        (short)0, acc[mi][ni], false, false);
        }
        __syncthreads();              // everyone done reading cur before it's re-staged
    }

    // epilogue: VGPR r holds row r + 8*hi, lane l15 holds the column
#pragma unroll
    for (int mi = 0; mi < 2; ++mi) {
#pragma unroll
        for (int ni = 0; ni < 4; ++ni) {
            int gcol = bn0 + wn0 + ni * 16 + l15;
#pragma unroll
            for (int r = 0; r < 8; ++r) {
                int grow = bm0 + wm0 + mi * 16 + hi * 8 + r;
                if (grow < M && gcol < Nc) {
                    float v = acc[mi][ni][r];
                    if (ZERO_DIAG && grow == gcol) v = 0.f;
                    C[(size_t)grow * ldc + gcol] = v;
                }
            }
        }
    }
}

extern "C" void kernel_launch(void* const* d_in, const int* in_sizes, int n_in,
                              void* d_out, int out_size, void* d_ws, size_t ws_size,
                              hipStream_t stream)
{
    (void)in_sizes; (void)n_in; (void)out_size; (void)ws_size;
    const int N = 10000, F = 512, H1 = 256, H2 = 128;

    const float* x   = (const float*)d_in[0];  // [N, F]
    const float* adj = (const float*)d_in[1];  // [N, N]
    const float* w1  = (const float*)d_in[2];  // [F, H1]
    const float* w2  = (const float*)d_in[3];  // [H1, H2]

    float* out = (float*)d_out;
    float* rec = out;                          // [N, N]   h1 @ h2^T
    float* h1  = rec + (size_t)N * N;          // [N, H1]
    float* h2  = h1 + (size_t)N * H1;          // [N, H2]
    float* a2  = h2 + (size_t)N * H2;          // [N, N]   adj@adj, diag zeroed

    float* t1 = (float*)d_ws;                  // [N, H1]  x @ w1
    float* t2 = t1 + (size_t)N * H1;           // [N, H2]  h1 @ w2

    dim3 blk(256);
    auto grid = [](int M, int Nc) { return dim3((Nc + TILE_N - 1) / TILE_N,
                                                (M + TILE_M - 1) / TILE_M); };

    // t1 = x @ w1
    wmma_gemm_f32<false, false><<<grid(N, H1), blk, 0, stream>>>(x,   w1, t1, N, H1, F,  F,  H1, H1);
    // h1 = adj @ t1
    wmma_gemm_f32<false, false><<<grid(N, H1), blk, 0, stream>>>(adj, t1, h1, N, H1, N,  N,  H1, H1);
    // t2 = h1 @ w2
    wmma_gemm_f32<false, false><<<grid(N, H2), blk, 0, stream>>>(h1,  w2, t2, N, H2, H1, H1, H2, H2);
    // h2 = adj @ t2
    wmma_gemm_f32<false, false><<<grid(N, H2), blk, 0, stream>>>(adj, t2, h2, N, H2, N,  N,  H2, H2);
    // a2 = adj @ adj with zeroed diagonal  (dominant 2-TFLOP GEMM)
    wmma_gemm_f32<false, true ><<<grid(N, N),  blk, 0, stream>>>(adj, adj, a2, N, N, N,  N,  N,  N);
    // rec = h1 @ h2^T (contract over the common 128 dim)
    wmma_gemm_f32<true,  false><<<grid(N, N),  blk, 0, stream>>>(h1,  h2, rec, N, N, H2, H1, H2, N);
}